// Embedding_2439541424257
// MI455X (gfx1250) — compile-verified
//
#include <hip/hip_runtime.h>
#include <hip/hip_bf16.h>
#include <math.h>

// ---------------- problem constants ----------------
#define NUM_PIECES   11
#define NUM_VEC      4096      // V
#define VEC_DIM      256       // D
#define NUM_FEAT     45056     // F = 11 * 4096
#define ROWS_PER_WG  64        // 64 | 4096 -> a block never crosses a piece
#define T1           256       // threads in streaming kernel (8 waves)

typedef __attribute__((ext_vector_type(2))) float v2f;
typedef __attribute__((ext_vector_type(8))) float v8f;

// ---------------- kernel 0: zero the padded A matrix (16 x 4096) ----------
__global__ void zero_kernel(float* __restrict__ p, int n) {
    int i = blockIdx.x * blockDim.x + threadIdx.x;
    if (i < n) p[i] = 0.0f;
}

// ---------------- kernel 1: streaming softmax + per-piece mass ------------
// One workgroup handles 64 consecutive feature rows (all in one piece).
// Each thread owns v = 4*t + 1024*i + j (i<4, j<4): float4-coalesced loads,
// row kept in registers, acc[] in LDS, one atomic flush per WG at the end.
__global__ void __launch_bounds__(T1)
softmax_mass_kernel(const float* __restrict__ logits,
                    const float* __restrict__ gumbel,
                    float* __restrict__ Apad /* 16 x 4096 */) {
    __shared__ float acc[NUM_VEC];        // 16 KB
    __shared__ float redA[8];
    __shared__ float redB[8];

    const int t    = threadIdx.x;
    const int lane = t & 31;
    const int wid  = t >> 5;

    // zero LDS accumulator
    float4* acc4 = (float4*)acc;
    #pragma unroll
    for (int i = 0; i < 4; ++i) acc4[t + T1 * i] = make_float4(0.f, 0.f, 0.f, 0.f);
    __syncthreads();

    const int f0    = blockIdx.x * ROWS_PER_WG;
    const int piece = f0 >> 12;                     // f0 / 4096

    for (int r = 0; r < ROWS_PER_WG; ++r) {
        const size_t rowoff = (size_t)(f0 + r) * NUM_VEC;
        const float4* lp = (const float4*)(logits + rowoff);
        const float4* gp = (const float4*)(gumbel + rowoff);

        float x[16];
        float mx = -INFINITY;
        #pragma unroll
        for (int i = 0; i < 4; ++i) {
            float4 lv = lp[t + T1 * i];
            float4 gv = gp[t + T1 * i];
            x[4*i+0] = lv.x + gv.x;  x[4*i+1] = lv.y + gv.y;
            x[4*i+2] = lv.z + gv.z;  x[4*i+3] = lv.w + gv.w;
            mx = fmaxf(mx, fmaxf(fmaxf(x[4*i], x[4*i+1]), fmaxf(x[4*i+2], x[4*i+3])));
        }
        // wave32 max reduce, then cross-wave via LDS
        #pragma unroll
        for (int off = 16; off >= 1; off >>= 1)
            mx = fmaxf(mx, __shfl_xor(mx, off, 32));
        if (lane == 0) redA[wid] = mx;
        __syncthreads();
        float rowmax = redA[0];
        #pragma unroll
        for (int w = 1; w < 8; ++w) rowmax = fmaxf(rowmax, redA[w]);

        float e[16];
        float s = 0.0f;
        #pragma unroll
        for (int j = 0; j < 16; ++j) { e[j] = __expf(x[j] - rowmax); s += e[j]; }
        #pragma unroll
        for (int off = 16; off >= 1; off >>= 1)
            s += __shfl_xor(s, off, 32);
        if (lane == 0) redB[wid] = s;
        __syncthreads();
        float rs = 0.0f;
        #pragma unroll
        for (int w = 0; w < 8; ++w) rs += redB[w];
        const float inv = 1.0f / rs;

        // accumulate probs into thread-private LDS slots (no conflicts/races)
        #pragma unroll
        for (int i = 0; i < 4; ++i) {
            float4 a = acc4[t + T1 * i];
            a.x += e[4*i+0] * inv;  a.y += e[4*i+1] * inv;
            a.z += e[4*i+2] * inv;  a.w += e[4*i+3] * inv;
            acc4[t + T1 * i] = a;
        }
    }
    __syncthreads();

    // flush piece-partial mass with native f32 atomics
    float* target = Apad + (size_t)piece * NUM_VEC;
    #pragma unroll
    for (int i = 0; i < 4; ++i) {
        #pragma unroll
        for (int j = 0; j < 4; ++j) {
            int v = 4 * t + 1024 * i + j;
            unsafeAtomicAdd(&target[v], acc[v]);
        }
    }
}

// ---------------- kernel 2: WMMA GEMM (16x4096 @ 4096x256) + roll-combine --
// 1 workgroup, 16 waves. Wave w computes D tile columns [16w, 16w+16) with
// V_WMMA_F32_16X16X4_F32 over K=4096, stores D into LDS, then threads 0..255
// produce out[d] = sum_p D[p][(d - 16p) mod 256].
__global__ void __launch_bounds__(512)
gemm_roll_kernel(const float* __restrict__ Apad,   // 16 x 4096
                 const float* __restrict__ base,   // 4096 x 256
                 float* __restrict__ out) {        // 256
    __shared__ float Dmat[16][VEC_DIM];            // 16 KB

    const int tid   = threadIdx.x;
    const int wid   = tid >> 5;      // N-tile index 0..15
    const int lane  = tid & 31;
    const int col16 = lane & 15;
    const int hi    = lane >> 4;     // 0: K pair {0,1}; 1: K pair {2,3}
    const int m     = col16;         // A-matrix row for this lane
    const int n     = wid * 16 + col16;  // global D column

    v8f c = {};                      // f32 accumulator, zero-init
    const float* arow = Apad + (size_t)m * NUM_VEC + 2 * hi;

    for (int k = 0; k < NUM_VEC; k += 4) {
        v2f a;
        a.x = arow[k];
        a.y = arow[k + 1];
        const int kk = k + 2 * hi;
        v2f b;
        b.x = base[(size_t)kk * VEC_DIM + n];
        b.y = base[(size_t)(kk + 1) * VEC_DIM + n];
        // D = A x B + C   (8 args: neg_a, A, neg_b, B, c_mod, C, reuse_a, reuse_b)
        c = __builtin_amdgcn_wmma_f32_16x16x4_f32(
                false, a, false, b, (short)0, c, false, false);
    }

    // C/D layout: VGPR j -> rows j (lanes 0-15) and j+8 (lanes 16-31)
    #pragma unroll
    for (int j = 0; j < 8; ++j) {
        Dmat[j + 8 * hi][n] = c[j];
    }
    __syncthreads();

    if (tid < VEC_DIM) {
        float s = 0.0f;
        #pragma unroll
        for (int p = 0; p < NUM_PIECES; ++p)
            s += Dmat[p][(tid - 16 * p) & (VEC_DIM - 1)];
        out[tid] = s;
    }
}

// ---------------- launch ----------------
extern "C" void kernel_launch(void* const* d_in, const int* in_sizes, int n_in,
                              void* d_out, int out_size, void* d_ws, size_t ws_size,
                              hipStream_t stream) {
    // input order: active_features, indices_logits, shared_vectors_base, gumbel_noise
    const float* logits = (const float*)d_in[1];   // (45056, 4096)
    const float* base   = (const float*)d_in[2];   // (4096, 256)
    const float* gumbel = (const float*)d_in[3];   // (1, 45056, 4096)
    float* out  = (float*)d_out;                   // (1, 256)
    float* Apad = (float*)d_ws;                    // 16 x 4096 f32 = 256 KB

    const int apad_elems = 16 * NUM_VEC;
    zero_kernel<<<(apad_elems + 255) / 256, 256, 0, stream>>>(Apad, apad_elems);

    softmax_mass_kernel<<<NUM_FEAT / ROWS_PER_WG, T1, 0, stream>>>(logits, gumbel, Apad);

    gemm_roll_kernel<<<1, 512, 0, stream>>>(Apad, base, out);
}